// YoloLoss_24610162606149
// MI455X (gfx1250) — compile-verified
//
#include <hip/hip_runtime.h>

typedef __attribute__((ext_vector_type(2))) float v2f;
typedef __attribute__((ext_vector_type(8))) float v8f;

#define LAMBDA_COORD 5.0f
#define LAMBDA_NOOBJ 0.5f

// Pass 1: one thread = one (b, i, j) cell. Streams 60 floats/cell (float2
// loads, 8B-aligned since cell stride = 120B), computes the weighted YOLO
// cell loss, reduces the wave via one V_WMMA_F32_16X16X4_F32 (B = ones),
// folds 8 wave sums per block through LDS, writes one partial per block.
__global__ __launch_bounds__(256) void yolo_pass1(const float* __restrict__ pred,
                                                  const float* __restrict__ tgt,
                                                  float* __restrict__ partials,
                                                  int nCells) {
    const int gid   = blockIdx.x * 256 + threadIdx.x;
    const int idx   = (gid < nCells) ? gid : (nCells - 1);   // clamp: keep EXEC full
    const float valid = (gid < nCells) ? 1.0f : 0.0f;

    const float2* __restrict__ p2 = reinterpret_cast<const float2*>(pred + (size_t)idx * 30);
    const float2* __restrict__ t2 = reinterpret_cast<const float2*>(tgt  + (size_t)idx * 30);

    float pv[30], tv[30];
#pragma unroll
    for (int i = 0; i < 15; ++i) {
        float2 a = p2[i]; pv[2 * i] = a.x; pv[2 * i + 1] = a.y;
        float2 b = t2[i]; tv[2 * i] = b.x; tv[2 * i + 1] = b.y;
    }

    // IoU of box0 = pred[0:4] and box1 = pred[5:9] vs tbox = target[0:4]
    const float t0 = tv[0], t1 = tv[1], t2c = tv[2], t3 = tv[3], t4 = tv[4];
    const float a2 = (t2c - t0) * (t3 - t1);

    float w0 = fmaxf(fminf(pv[2], t2c) - fmaxf(pv[0], t0), 0.0f);
    float h0 = fmaxf(fminf(pv[3], t3)  - fmaxf(pv[1], t1), 0.0f);
    float inter0 = w0 * h0;
    float a1_0 = (pv[2] - pv[0]) * (pv[3] - pv[1]);
    float iou0 = inter0 / (a1_0 + a2 - inter0);

    float w1 = fmaxf(fminf(pv[7], t2c) - fmaxf(pv[5], t0), 0.0f);
    float h1 = fmaxf(fminf(pv[8], t3)  - fmaxf(pv[6], t1), 0.0f);
    float inter1 = w1 * h1;
    float a1_1 = (pv[7] - pv[5]) * (pv[8] - pv[6]);
    float iou1 = inter1 / (a1_1 + a2 - inter1);

    const bool pick0 = (iou0 >= iou1);          // argmax over 2, first-wins tie
    const float c0 = pick0 ? pv[0] : pv[5];
    const float c1 = pick0 ? pv[1] : pv[6];
    const float c2 = pick0 ? pv[2] : pv[7];
    const float c3 = pick0 ? pv[3] : pv[8];

    const bool  obj  = (t4 > 0.0f);
    const float objf = obj ? 1.0f : 0.0f;

    // loss1: xy
    const float xy0 = obj ? c0 : 0.0f;
    const float xy1 = obj ? c1 : 0.0f;
    const float d0 = xy0 - t0, d1 = xy1 - t1;
    const float l1 = d0 * d0 + d1 * d1;

    // loss2: sqrt wh
    const float sb0 = sqrtf(obj ? c2 : 1.0f);
    const float sb1 = sqrtf(obj ? c3 : 1.0f);
    const float st0 = sqrtf(t2c);
    const float st1 = sqrtf(t3);
    const float e0 = sb0 - st0, e1 = sb1 - st1;
    const float l2 = objf * (e0 * e0 + e1 * e1)
                   + (1.0f - objf) * (st0 * st0 + st1 * st1);

    // loss3 / loss4: confidence
    const float conf = pv[4];
    const float v3 = (obj ? conf : 0.0f) - t4;
    const float l3 = v3 * v3;
    const float v4 = (obj ? 1.0f : conf) - t4;
    const float l4 = v4 * v4;

    // loss5: classes 10..29
    float l5 = 0.0f;
#pragma unroll
    for (int c = 10; c < 30; ++c) {
        const float vc = (obj ? pv[c] : 0.0f) - tv[c];
        l5 = fmaf(vc, vc, l5);
    }

    const float cell = valid * (LAMBDA_COORD * (l1 + l2) + l3 + LAMBDA_NOOBJ * l4 + l5);

    // ---- wave32 reduction via WMMA (ones-matrix row-sum trick) ----
    // A(16x4): lane m VGPR0 -> A[m][0]=cell_m ; lane m+16 VGPR0 -> A[m][2]=cell_{m+16}
    // B(4x16) = ones  =>  D[m][n] = cell_m + cell_{m+16}
    v2f a; a[0] = cell; a[1] = 0.0f;
    v2f b; b[0] = 1.0f; b[1] = 1.0f;
    v8f cacc = {};
    v8f d = __builtin_amdgcn_wmma_f32_16x16x4_f32(
        /*neg_a=*/false, a, /*neg_b=*/false, b,
        /*c_mod=*/(short)0, cacc, /*reuse_a=*/false, /*reuse_b=*/false);
    // lanes 0-15 hold rows 0-7, lanes 16-31 hold rows 8-15
    float s = d[0] + d[1] + d[2] + d[3] + d[4] + d[5] + d[6] + d[7];

    __shared__ float lsum[16];
    const int lane = threadIdx.x & 31;
    const int wid  = threadIdx.x >> 5;
    if (lane == 0)  lsum[2 * wid + 0] = s;   // rows 0-7 half-sum
    if (lane == 16) lsum[2 * wid + 1] = s;   // rows 8-15 half-sum
    __syncthreads();
    if (threadIdx.x == 0) {
        float bs = 0.0f;
#pragma unroll
        for (int i = 0; i < 16; ++i) bs += lsum[i];
        partials[blockIdx.x] = bs;
    }
}

// Pass 2: single block folds block partials deterministically and scales.
__global__ __launch_bounds__(256) void yolo_pass2(const float* __restrict__ partials,
                                                  float* __restrict__ out,
                                                  int nPartials, float invB) {
    __shared__ float sdata[256];
    float s = 0.0f;
    for (int i = threadIdx.x; i < nPartials; i += 256) s += partials[i];
    sdata[threadIdx.x] = s;
    __syncthreads();
#pragma unroll
    for (int stride = 128; stride > 0; stride >>= 1) {
        if (threadIdx.x < (unsigned)stride) sdata[threadIdx.x] += sdata[threadIdx.x + stride];
        __syncthreads();
    }
    if (threadIdx.x == 0) out[0] = sdata[0] * invB;
}

extern "C" void kernel_launch(void* const* d_in, const int* in_sizes, int n_in,
                              void* d_out, int out_size, void* d_ws, size_t ws_size,
                              hipStream_t stream) {
    const float* pred = (const float*)d_in[0];
    const float* tgt  = (const float*)d_in[1];
    float* out        = (float*)d_out;
    float* partials   = (float*)d_ws;

    const int nCells = in_sizes[0] / 30;                 // 16384*7*7 = 802816
    const int blocks = (nCells + 255) / 256;             // 3136 (exact)
    const float invB = 49.0f / (float)nCells;            // 1/B = 1/16384 (exact)

    yolo_pass1<<<blocks, 256, 0, stream>>>(pred, tgt, partials, nCells);
    yolo_pass2<<<1, 256, 0, stream>>>(partials, out, blocks, invB);
}